// GAT_3083786518794
// MI455X (gfx1250) — compile-verified
//
#include <hip/hip_runtime.h>
#include <math.h>

// ---------------------------------------------------------------------------
// 2-layer GAT, N=100000. GEMMs via V_WMMA_F32_16X16X4_F32 (fp32 precision,
// matrix pipe). Edge softmax/aggregation via atomics (memory-roofline bound:
// ~3.5GB gather + ~0.5GB atomic traffic for layer 2 -> ~200-400us @ 23.3TB/s).
// ---------------------------------------------------------------------------

typedef __attribute__((ext_vector_type(2))) float v2f;
typedef __attribute__((ext_vector_type(8))) float v8f;

#define NNODES 100000
#define NEG_SLOPE 0.2f

// ---------- fp32 WMMA: one wave computes a 16x16 tile of C = A@B ----------
__device__ __forceinline__ void wmma_tile_f32(const float* __restrict__ A, int lda,
                                              const float* __restrict__ B, int ldb,
                                              float* __restrict__ C, int ldc,
                                              int K, int mBase, int nBase) {
    const int lane = threadIdx.x & 31;
    const int mn   = lane & 15;   // A row / B col / C col
    const int kh   = lane >> 4;   // K half: 0 -> K=0,1 ; 1 -> K=2,3
    v8f c = {};
    for (int k0 = 0; k0 < K; k0 += 4) {
        const float* ap = A + (mBase + mn) * lda + k0 + 2 * kh;
        v2f a; a.x = ap[0]; a.y = ap[1];
        const float* bp = B + (k0 + 2 * kh) * ldb + nBase + mn;
        v2f b; b.x = bp[0]; b.y = bp[ldb];
        c = __builtin_amdgcn_wmma_f32_16x16x4_f32(false, a, false, b,
                                                  (short)0, c, false, false);
    }
#pragma unroll
    for (int r = 0; r < 8; ++r) {
        int row = mBase + r + 8 * kh;           // VGPR r: row r (lanes 0-15) / r+8
        C[row * ldc + nBase + mn] = c[r];
    }
}

// GEMM1: h1[N,64] = x[N,256] @ W1[256,64].  128 thr = 4 waves = 4 col tiles.
__global__ void gat_gemm1(const float* __restrict__ x, const float* __restrict__ W1,
                          float* __restrict__ h1) {
    const int wave  = threadIdx.x >> 5;
    const int mBase = blockIdx.x * 16;          // N = 6250 * 16 exactly
    wmma_tile_f32(x, 256, W1, 64, h1, 64, 256, mBase, wave * 16);
}

// GEMM2: h2[N,512] = act[N,64] @ W2[64,512]. 256 thr = 8 waves; grid.x=4 col blks.
__global__ void gat_gemm2(const float* __restrict__ act, const float* __restrict__ W2,
                          float* __restrict__ h2) {
    const int wave  = threadIdx.x >> 5;
    const int mBase = blockIdx.y * 16;
    const int nBase = (blockIdx.x * 8 + wave) * 16;
    wmma_tile_f32(act, 64, W2, 512, h2, 512, 64, mBase, nBase);
}

// ---------- per-(node,head) attention coefficients ----------
__global__ void gat_alpha(const float* __restrict__ h, const float* __restrict__ a_src,
                          const float* __restrict__ a_dst, float* __restrict__ as_out,
                          float* __restrict__ ad_out, int F) {
    int idx = blockIdx.x * blockDim.x + threadIdx.x;
    if (idx >= NNODES * 8) return;
    const int n = idx >> 3, head = idx & 7;
    const float* hp = h + (n * 8 + head) * F;
    const float* sp = a_src + head * F;
    const float* dp = a_dst + head * F;
    float s = 0.f, d = 0.f;
    for (int f = 0; f < F; ++f) { float v = hp[f]; s += v * sp[f]; d += v * dp[f]; }
    as_out[idx] = s;
    ad_out[idx] = d;
}

// ---------- edge helpers ----------
__device__ __forceinline__ void edge_sd(const int* __restrict__ e0, const int* __restrict__ e1,
                                        int E0, int i, int& s, int& d) {
    if (i < E0) { s = e0[i]; d = e1[i]; }
    else        { s = d = i - E0; }            // appended self-loops
}
__device__ __forceinline__ int   f2ord(float f) { int i = __float_as_int(f); return i >= 0 ? i : (i ^ 0x7FFFFFFF); }
__device__ __forceinline__ float ord2f(int i)   { return __int_as_float(i >= 0 ? i : (i ^ 0x7FFFFFFF)); }
__device__ __forceinline__ float leaky(float e) { return e > 0.f ? e : NEG_SLOPE * e; }

__global__ void gat_edge_max(const int* __restrict__ e0, const int* __restrict__ e1,
                             int E0, int Etot, const float* __restrict__ as,
                             const float* __restrict__ ad, int* __restrict__ m) {
    int idx = blockIdx.x * blockDim.x + threadIdx.x;
    if (idx >= Etot * 8) return;
    int ei = idx >> 3, hd = idx & 7, s, d;
    edge_sd(e0, e1, E0, ei, s, d);
    float e = leaky(as[s * 8 + hd] + ad[d * 8 + hd]);
    atomicMax(&m[d * 8 + hd], f2ord(e));
}

__global__ void gat_edge_sum(const int* __restrict__ e0, const int* __restrict__ e1,
                             int E0, int Etot, const float* __restrict__ as,
                             const float* __restrict__ ad, const int* __restrict__ m,
                             float* __restrict__ den) {
    int idx = blockIdx.x * blockDim.x + threadIdx.x;
    if (idx >= Etot * 8) return;
    int ei = idx >> 3, hd = idx & 7, s, d;
    edge_sd(e0, e1, E0, ei, s, d);
    float e = leaky(as[s * 8 + hd] + ad[d * 8 + hd]);
    atomicAdd(&den[d * 8 + hd], expf(e - ord2f(m[d * 8 + hd])));
}

// scatter: agg[(dst,head),0:F] += alpha * h[(src,head),0:F]
__global__ void gat_edge_agg(const int* __restrict__ e0, const int* __restrict__ e1,
                             int E0, int Etot, const float* __restrict__ as,
                             const float* __restrict__ ad, const int* __restrict__ m,
                             const float* __restrict__ den, const float* __restrict__ h,
                             float* __restrict__ agg, int F) {
    int idx = blockIdx.x * blockDim.x + threadIdx.x;
    if (idx >= Etot * 8) return;
    int ei = idx >> 3, hd = idx & 7, s, d;
    edge_sd(e0, e1, E0, ei, s, d);
    float e = leaky(as[s * 8 + hd] + ad[d * 8 + hd]);
    float alpha = expf(e - ord2f(m[d * 8 + hd])) / den[d * 8 + hd];
    const float* hp = h   + (size_t)(s * 8 + hd) * F;
    float*       op = agg + (size_t)(d * 8 + hd) * F;
    for (int f = 0; f < F; f += 4) {
        float4 v = *(const float4*)(hp + f);
        atomicAdd(op + f + 0, alpha * v.x);
        atomicAdd(op + f + 1, alpha * v.y);
        atomicAdd(op + f + 2, alpha * v.z);
        atomicAdd(op + f + 3, alpha * v.w);
    }
}

// ---------- elementwise ----------
__global__ void gat_elu_bias(float* __restrict__ a, const float* __restrict__ b1) {
    int idx = blockIdx.x * blockDim.x + threadIdx.x;
    if (idx >= NNODES * 64) return;
    float v = a[idx] + b1[idx & 63];
    a[idx] = v > 0.f ? v : expm1f(v);           // ELU
}

__global__ void gat_mean_bias(const float* __restrict__ agg2, const float* __restrict__ b2,
                              float* __restrict__ out) {
    int idx = blockIdx.x * blockDim.x + threadIdx.x;
    if (idx >= NNODES * 64) return;
    int n = idx >> 6, f = idx & 63;
    float s = 0.f;
#pragma unroll
    for (int hd = 0; hd < 8; ++hd) s += agg2[(size_t)(n * 8 + hd) * 64 + f];
    out[idx] = s * 0.125f + b2[f];
}

__global__ void gat_init_m(int* __restrict__ m1, int* __restrict__ m2) {
    int idx = blockIdx.x * blockDim.x + threadIdx.x;
    if (idx >= NNODES * 8) return;
    m1[idx] = (int)0x80000000;                  // <= ordered(-inf)
    m2[idx] = (int)0x80000000;
}

// ---------------------------------------------------------------------------
extern "C" void kernel_launch(void* const* d_in, const int* in_sizes, int n_in,
                              void* d_out, int out_size, void* d_ws, size_t ws_size,
                              hipStream_t stream) {
    const float* x      = (const float*)d_in[0];
    const float* W1     = (const float*)d_in[1];
    const float* a_src1 = (const float*)d_in[2];
    const float* a_dst1 = (const float*)d_in[3];
    const float* b1     = (const float*)d_in[4];
    const float* W2     = (const float*)d_in[5];
    const float* a_src2 = (const float*)d_in[6];
    const float* a_dst2 = (const float*)d_in[7];
    const float* b2     = (const float*)d_in[8];
    const int*   edge   = (const int*)d_in[9];

    const int E0   = in_sizes[9] / 2;           // 1,600,000
    const int Etot = E0 + NNODES;               // + self loops
    const int* e0 = edge;
    const int* e1 = edge + E0;

    // workspace layout (floats)
    float* h1   = (float*)d_ws;                 // N*64
    float* as1  = h1   + (size_t)NNODES * 64;   // N*8
    float* ad1  = as1  + (size_t)NNODES * 8;
    int*   m1   = (int*)(ad1 + (size_t)NNODES * 8);
    float* den1 = (float*)(m1 + (size_t)NNODES * 8);
    float* agg1 = den1 + (size_t)NNODES * 8;    // N*64 (becomes act after ELU)
    float* h2   = agg1 + (size_t)NNODES * 64;   // N*512
    float* as2  = h2   + (size_t)NNODES * 512;
    float* ad2  = as2  + (size_t)NNODES * 8;
    int*   m2   = (int*)(ad2 + (size_t)NNODES * 8);
    float* den2 = (float*)(m2 + (size_t)NNODES * 8);
    float* agg2 = den2 + (size_t)NNODES * 8;    // N*512

    const int T = 256;
    const int gNH  = (NNODES * 8 + T - 1) / T;
    const int gNF  = (NNODES * 64 + T - 1) / T;
    const int gEH  = (Etot * 8 + T - 1) / T;

    // per-call re-init (harness does not re-poison between graph replays)
    hipMemsetAsync(den1, 0, (size_t)NNODES * 8   * sizeof(float), stream);
    hipMemsetAsync(den2, 0, (size_t)NNODES * 8   * sizeof(float), stream);
    hipMemsetAsync(agg1, 0, (size_t)NNODES * 64  * sizeof(float), stream);
    hipMemsetAsync(agg2, 0, (size_t)NNODES * 512 * sizeof(float), stream);
    gat_init_m<<<gNH, T, 0, stream>>>(m1, m2);

    // ---- layer 1 ----
    gat_gemm1<<<NNODES / 16, 128, 0, stream>>>(x, W1, h1);
    gat_alpha<<<gNH, T, 0, stream>>>(h1, a_src1, a_dst1, as1, ad1, 8);
    gat_edge_max<<<gEH, T, 0, stream>>>(e0, e1, E0, Etot, as1, ad1, m1);
    gat_edge_sum<<<gEH, T, 0, stream>>>(e0, e1, E0, Etot, as1, ad1, m1, den1);
    gat_edge_agg<<<gEH, T, 0, stream>>>(e0, e1, E0, Etot, as1, ad1, m1, den1, h1, agg1, 8);
    gat_elu_bias<<<gNF, T, 0, stream>>>(agg1, b1);

    // ---- layer 2 ----
    gat_gemm2<<<dim3(4, NNODES / 16), 256, 0, stream>>>(agg1, W2, h2);
    gat_alpha<<<gNH, T, 0, stream>>>(h2, a_src2, a_dst2, as2, ad2, 64);
    gat_edge_max<<<gEH, T, 0, stream>>>(e0, e1, E0, Etot, as2, ad2, m2);
    gat_edge_sum<<<gEH, T, 0, stream>>>(e0, e1, E0, Etot, as2, ad2, m2, den2);
    gat_edge_agg<<<gEH, T, 0, stream>>>(e0, e1, E0, Etot, as2, ad2, m2, den2, h2, agg2, 64);
    gat_mean_bias<<<gNF, T, 0, stream>>>(agg2, b2, (float*)d_out);
}